// msgFunc_EN_8598524527134
// MI455X (gfx1250) — compile-verified
//
#include <hip/hip_runtime.h>

typedef __attribute__((ext_vector_type(2))) float v2f;
typedef __attribute__((ext_vector_type(8))) float v8f;

#define BB 4
#define NN 128
#define DD 64
#define ISPLIT 4                 // i-range split across wave groups
#define NWAVES (4 * ISPLIT)      // 4 k-tiles x ISPLIT i-slices = 16 waves

// out[b,j,k] = (1/NN) * sum_{i,l} A[b,i,j,k,l] * h[b,i,l]
// One workgroup per (b,j), 16 waves. Wave = (ihalf, ktile): accumulates the
// 16x64 matvec partial over i in {ihalf, ihalf+4, ...} via WMMA f32 16x16x4
// (B-operand = h broadcast into all 16 columns -> D columns replicated).
// Deterministic LDS reduction combines the ISPLIT partials per k.
__global__ __launch_bounds__(32 * NWAVES) void msgfunc_wmma_kernel(
    const float* __restrict__ A, const float* __restrict__ h,
    float* __restrict__ out) {
  __shared__ float partial[NWAVES * 16];   // [wave][16 k-rows]

  const int bj = blockIdx.x;             // 0 .. BB*NN-1
  const int b  = bj >> 7;                // / NN
  const int j  = bj & (NN - 1);
  const int wave = threadIdx.x >> 5;
  const int lane = threadIdx.x & 31;
  const int kt = wave & 3;               // k-tile (0..3)
  const int ih = wave >> 2;              // i-slice (0..ISPLIT-1)
  const int m    = lane & 15;            // M row (A) / N col (B) within tile
  const int half = lane >> 4;            // selects K+2 per ISA 16x4 f32 layout

  // A[b,i,j,k,l]: l stride 1, k stride DD, j stride DD*DD, i stride NN*DD*DD
  const long long iStrideA = (long long)NN * DD * DD;
  const float* Abase = A + ((long long)b * NN * NN + j) * (DD * DD)
                         + (long long)(kt * 16 + m) * DD + 2 * half;
  const float* hbase = h + (long long)b * NN * DD + 2 * half;

  v8f acc = {};
  for (int i = ih; i < NN; i += ISPLIT) {
    const float* ap = Abase + (long long)i * iStrideA;
    const float* hp = hbase + i * DD;
#pragma unroll
    for (int c = 0; c < DD; c += 4) {
      // A fragment: VGPR v holds A[k, c + 2*half + v]; streamed once -> NT
      v2f afrag = __builtin_nontemporal_load((const v2f*)(ap + c));
      // B fragment: VGPR v holds h[c + 2*half + v], same for all 16 columns
      v2f bfrag = *(const v2f*)(hp + c);
      acc = __builtin_amdgcn_wmma_f32_16x16x4_f32(
          /*neg_a=*/false, afrag, /*neg_b=*/false, bfrag,
          /*c_mod=*/(short)0, acc, /*reuse_a=*/false, /*reuse_b=*/false);
    }
  }

  // D layout: VGPR p -> M = p (lanes 0-15) or p+8 (lanes 16-31); columns are
  // replicated, so lanes 0 and 16 carry the 16-row partial for this wave.
  if (lane == 0) {
#pragma unroll
    for (int p = 0; p < 8; ++p) partial[wave * 16 + p] = acc[p];
  } else if (lane == 16) {
#pragma unroll
    for (int p = 0; p < 8; ++p) partial[wave * 16 + 8 + p] = acc[p];
  }
  __syncthreads();

  // partial[ih*64 + k] holds slice ih's contribution to row k (k = kt*16+p).
  // Fixed-order 4-way sum -> bit-deterministic output.
  const int t = threadIdx.x;
  if (t < DD) {
    float s = ((partial[t] + partial[64 + t]) +
               (partial[128 + t] + partial[192 + t]));
    out[((long long)b * NN + j) * DD + t] = s * (1.0f / (float)NN);
  }
}

extern "C" void kernel_launch(void* const* d_in, const int* in_sizes, int n_in,
                              void* d_out, int out_size, void* d_ws, size_t ws_size,
                              hipStream_t stream) {
  const float* h = (const float*)d_in[0];   // [BB, NN, DD]
  const float* A = (const float*)d_in[1];   // [BB, NN, NN, DD, DD]
  float* out = (float*)d_out;               // [BB, NN, DD]
  (void)in_sizes; (void)n_in; (void)out_size; (void)d_ws; (void)ws_size;

  msgfunc_wmma_kernel<<<dim3(BB * NN), dim3(32 * NWAVES), 0, stream>>>(A, h, out);
}